// GINO_51075751084703
// MI455X (gfx1250) — compile-verified
//
#include <hip/hip_runtime.h>
#include <hip/hip_bf16.h>

#define B_    4
#define CIN   3
#define N_    8192
#define W_    128
#define MODES 32
#define K2    64
#define DEG   16
#define NT    32768

typedef __attribute__((ext_vector_type(2))) float v2f;
typedef __attribute__((ext_vector_type(8))) float v8f;

__device__ __forceinline__ float gelu_f(float v) {
    return 0.5f * v * (1.0f + erff(v * 0.70710678118654752f));
}

// ---------------------------------------------------------------- basis table
// basis[n*64 + k] = cos(2*pi*k*n/N)   for k < 32
//                 = sin(2*pi*(k-32)*n/N) for k >= 32
__global__ void k_basis(float* basis) {
    int idx = blockIdx.x * 256 + threadIdx.x;
    if (idx >= N_ * K2) return;
    int n = idx / K2, k = idx % K2;
    int kk = k & 31;
    long long mm = ((long long)kk * n) % N_;
    double ang = 6.283185307179586476925287 * (double)mm / (double)N_;
    basis[idx] = (k < MODES) ? (float)cos(ang) : (float)sin(ang);
}

__global__ void k_zero(float* p, int n) {
    int i = blockIdx.x * 256 + threadIdx.x;
    if (i < n) p[i] = 0.0f;
}

// ------------------------------------------------------- encoder edge + mean
// mean_e [W xi; xj-xi] + b  ==  W [xi ; mean(xj)-xi] + b  (linear in xj)
__global__ void __launch_bounds__(128) k_enc_edge(
    const float* __restrict__ x, const int* __restrict__ src,
    const float* __restrict__ ew, const float* __restrict__ eb,
    float* __restrict__ h)
{
    int t = blockIdx.x;
    int o = threadIdx.x;
    __shared__ float sm[3];
    int b = t / N_, n = t % N_;
    if (o < 3) sm[o] = 0.0f;
    __syncthreads();
    if (o < 51) {  // 17 neighbors (16 random + self) x 3 channels
        int j = o / 3, c = o % 3;
        int s = (j < DEG) ? src[t * DEG + j] : t;
        int sb = s / N_, sn = s % N_;
        atomicAdd(&sm[c], x[((size_t)sb * CIN + c) * N_ + sn]);
    }
    __syncthreads();
    float xi0 = x[((size_t)b * CIN + 0) * N_ + n];
    float xi1 = x[((size_t)b * CIN + 1) * N_ + n];
    float xi2 = x[((size_t)b * CIN + 2) * N_ + n];
    const float inv = 1.0f / 17.0f;
    float z3 = sm[0] * inv - xi0;
    float z4 = sm[1] * inv - xi1;
    float z5 = sm[2] * inv - xi2;
    const float* w = ew + o * 6;
    float m = w[0]*xi0 + w[1]*xi1 + w[2]*xi2 + w[3]*z3 + w[4]*z4 + w[5]*z5 + eb[o];
    h[(size_t)t * W_ + o] = gelu_f(m);
}

// ------------------------------------------------------- gather-mean (17 nbrs)
__global__ void k_gather(const float* __restrict__ in, const int* __restrict__ src,
                         float* __restrict__ out, int C, int dogelu)
{
    int t = blockIdx.x;
    int c = threadIdx.x;
    float s = in[(size_t)t * C + c];  // self loop
    for (int j = 0; j < DEG; j++) {
        int si = src[t * DEG + j];
        s += in[(size_t)si * C + c];
    }
    s *= (1.0f / 17.0f);
    out[(size_t)t * C + c] = dogelu ? gelu_f(s) : s;
}

// ------------------------------------------------------- instance-norm stats
// accumulate per-(b,channel) sum & sumsq over n (coalesced, atomic combine)
__global__ void __launch_bounds__(128) k_stats_acc(
    const float* __restrict__ v, float* __restrict__ sums)
{
    int b = blockIdx.x;        // 0..3
    int chunk = blockIdx.y;    // 0..63
    int c = threadIdx.x;       // 0..127
    int n0 = chunk * (N_ / 64);
    const float* base = v + ((size_t)b * N_ + n0) * W_ + c;
    float s = 0.0f, q = 0.0f;
    for (int i = 0; i < N_ / 64; i++) {
        float x = base[(size_t)i * W_];
        s += x; q += x * x;
    }
    atomicAdd(&sums[b * W_ + c], s);
    atomicAdd(&sums[512 + b * W_ + c], q);
}

__global__ void k_stats_fin(const float* __restrict__ sums, float* __restrict__ stats) {
    int i = blockIdx.x * 128 + threadIdx.x;
    if (i >= 512) return;
    float m = sums[i] / (float)N_;
    float var = sums[512 + i] / (float)N_ - m * m;
    stats[i] = m;
    stats[512 + i] = rsqrtf(var + 1e-5f);
}

// -------------------------------------------- generic WMMA fp32 GEMM (K=128)
// C[r, o] = act( sum_i A[r,i]*Wt[o,i] + bias[o] (+ add[r,o]) )
// Optional fused instance-norm on A-load: a = (a - mean[b*128+i]) * rstd[...]
__global__ void __launch_bounds__(256) k_gemm(
    const float* __restrict__ A, const float* __restrict__ Wt,
    const float* __restrict__ bias, const float* __restrict__ add,
    const float* __restrict__ stats, float* __restrict__ C,
    int ldc, int actGelu)
{
    __shared__ float As[16 * 132];
    int rowbase = blockIdx.x * 16;
    int lane = threadIdx.x & 31;
    int wave = threadIdx.x >> 5;
    int colbase = blockIdx.y * 128 + wave * 16;

    for (int idx = threadIdx.x; idx < 2048; idx += 256) {
        int r = idx >> 7, c = idx & 127;
        size_t gr = (size_t)(rowbase + r);
        float v = A[gr * 128 + c];
        if (stats) {
            int b = (int)(gr / N_);
            v = (v - stats[b * 128 + c]) * stats[512 + b * 128 + c];
        }
        As[r * 132 + c] = v;
    }
    __syncthreads();

    int m = lane & 15;
    int kq = (lane >> 4) * 2;
    const float* wrow = Wt + (size_t)(colbase + m) * 128;
    v8f acc = {0.f, 0.f, 0.f, 0.f, 0.f, 0.f, 0.f, 0.f};
    for (int k = 0; k < 128; k += 4) {
        int ka = k + kq;
        v2f a = { As[m * 132 + ka], As[m * 132 + ka + 1] };
        v2f bf = { wrow[ka], wrow[ka + 1] };
        acc = __builtin_amdgcn_wmma_f32_16x16x4_f32(false, a, false, bf,
                                                    (short)0, acc, false, false);
    }

    int nglob = colbase + m;
    float bv = bias ? bias[nglob] : 0.0f;
    int mbase = (lane >> 4) * 8;
    for (int i = 0; i < 8; i++) {
        float v = acc[i] + bv;
        size_t off = (size_t)(rowbase + mbase + i) * ldc + nglob;
        if (add) v += add[off];
        if (actGelu) v = gelu_f(v);
        C[off] = v;
    }
}

// ------------------------------------------ truncated DFT (32 modes) via WMMA
// cc[r*64 + k]    = sum_n norm(h)[r,n] * cos(2pi k n / N)   (k<32)
// cc[r*64 + 32+k] = sum_n norm(h)[r,n] * sin(2pi k n / N)
// rows r = b*128 + channel.  K-split over grid.y, atomic combine.
__global__ void __launch_bounds__(128) k_dft(
    const float* __restrict__ h, const float* __restrict__ stats,
    const float* __restrict__ basis, float* __restrict__ cc)
{
    __shared__ float As[16 * 260];
    int rowbase = blockIdx.x * 16;          // 32 row tiles of (b,channel)
    int b = rowbase >> 7;
    int ibase = rowbase & 127;
    int kbase = blockIdx.y * 256;           // 32 K-splits

    for (int idx = threadIdx.x; idx < 16 * 256; idx += 128) {
        int rr = idx & 15, kk = idx >> 4;
        int r = rowbase + rr;
        float v = h[((size_t)b * N_ + kbase + kk) * 128 + ibase + rr];
        v = (v - stats[r]) * stats[512 + r];
        As[rr * 260 + kk] = v;
    }
    __syncthreads();

    int lane = threadIdx.x & 31, wave = threadIdx.x >> 5;
    int m = lane & 15;
    int kq = (lane >> 4) * 2;
    int col = wave * 16 + m;                // 0..63
    v8f acc = {0.f, 0.f, 0.f, 0.f, 0.f, 0.f, 0.f, 0.f};
    for (int k = 0; k < 256; k += 4) {
        int ka = k + kq;
        v2f a  = { As[m * 260 + ka], As[m * 260 + ka + 1] };
        v2f bf = { basis[(size_t)(kbase + ka) * K2 + col],
                   basis[(size_t)(kbase + ka + 1) * K2 + col] };
        acc = __builtin_amdgcn_wmma_f32_16x16x4_f32(false, a, false, bf,
                                                    (short)0, acc, false, false);
    }
    int mbase = (lane >> 4) * 8;
    for (int i = 0; i < 8; i++)
        atomicAdd(&cc[(size_t)(rowbase + mbase + i) * K2 + col], acc[i]);
}

// --------------------------------- spectral mix + fold irfft scaling factors
// vf = cr + i*ci with cr=cc[k], ci=-cc[32+k];  of = vf * (wr + i*wi)
// irfft coeffs:  A_k = f_k*Re(of),  B_k = -f_k*Im(of),  f_k = (k?2:1)/N
__global__ void __launch_bounds__(128) k_spectral(
    const float* __restrict__ cc, const float* __restrict__ wr,
    const float* __restrict__ wi, float* __restrict__ co)
{
    int bk = blockIdx.x;          // B*MODES = 128
    int b = bk >> 5, k = bk & 31;
    int o = threadIdx.x;
    __shared__ float scr[128], sci[128];
    scr[o] =  cc[(size_t)(b * 128 + o) * K2 + k];
    sci[o] = -cc[(size_t)(b * 128 + o) * K2 + 32 + k];
    __syncthreads();
    float orr = 0.0f, oii = 0.0f;
    for (int i = 0; i < 128; i++) {
        float wrv = wr[((size_t)i * 128 + o) * MODES + k];
        float wiv = wi[((size_t)i * 128 + o) * MODES + k];
        orr += scr[i] * wrv - sci[i] * wiv;
        oii += scr[i] * wiv + sci[i] * wrv;
    }
    float f = (k == 0 ? 1.0f : 2.0f) / (float)N_;
    co[(size_t)(b * 128 + o) * K2 + k]      =  f * orr;
    co[(size_t)(b * 128 + o) * K2 + 32 + k] = -f * oii;
}

// ------------------------------------------------ truncated iDFT via WMMA
// g[(b*N+n)*128 + o] = sum_k co[(b*128+o)*64+k] * basis[n*64+k]
__global__ void __launch_bounds__(256) k_idft(
    const float* __restrict__ co, const float* __restrict__ basis,
    float* __restrict__ g)
{
    int nbase = blockIdx.x * 16;   // 512 tiles over n
    int b = blockIdx.y;
    int lane = threadIdx.x & 31, wave = threadIdx.x >> 5;
    int m = lane & 15;
    int kq = (lane >> 4) * 2;
    int obase = wave * 16;
    v8f acc = {0.f, 0.f, 0.f, 0.f, 0.f, 0.f, 0.f, 0.f};
    const float* arow = basis + (size_t)(nbase + m) * K2;
    const float* brow = co + (size_t)(b * 128 + obase + m) * K2;
    for (int k = 0; k < K2; k += 4) {
        int ka = k + kq;
        v2f a  = { arow[ka], arow[ka + 1] };
        v2f bf = { brow[ka], brow[ka + 1] };
        acc = __builtin_amdgcn_wmma_f32_16x16x4_f32(false, a, false, bf,
                                                    (short)0, acc, false, false);
    }
    int mbase = (lane >> 4) * 8;
    for (int i = 0; i < 8; i++)
        g[((size_t)b * N_ + nbase + mbase + i) * 128 + obase + m] = acc[i];
}

// ---------------------------------------------------- decoder head 256 -> 1
__global__ void __launch_bounds__(256) k_matvec(
    const float* __restrict__ hdec, const float* __restrict__ w,
    const float* __restrict__ bptr, float* __restrict__ out)
{
    int t = blockIdx.x * 8 + (threadIdx.x >> 5);
    int lane = threadIdx.x & 31;
    const float* row = hdec + (size_t)t * 256;
    float s = 0.0f;
    for (int i = lane; i < 256; i += 32) s += row[i] * w[i];
    for (int d = 16; d; d >>= 1) s += __shfl_xor(s, d, 32);
    if (lane == 0) out[t] = s + bptr[0];
}

__global__ void k_gather1(const float* __restrict__ hl2, const int* __restrict__ src,
                          float* __restrict__ out)
{
    int t = blockIdx.x * 256 + threadIdx.x;
    if (t >= NT) return;
    float s = hl2[t];
    for (int j = 0; j < DEG; j++) s += hl2[src[t * DEG + j]];
    out[t] = s * (1.0f / 17.0f);
}

// ============================================================================
extern "C" void kernel_launch(void* const* d_in, const int* in_sizes, int n_in,
                              void* d_out, int out_size, void* d_ws, size_t ws_size,
                              hipStream_t stream)
{
    (void)in_sizes; (void)n_in; (void)out_size; (void)ws_size;

    const float* x       = (const float*)d_in[0];
    const int*   src     = (const int*)d_in[1];        // row 0 of edge_index
    const float* enc_e_w = (const float*)d_in[2];
    const float* enc_e_b = (const float*)d_in[3];
    const float* enc_c_w = (const float*)d_in[4];
    const float* enc_c_b = (const float*)d_in[5];
    const float* wr      = (const float*)d_in[6];
    const float* wi      = (const float*)d_in[7];
    const float* w_w     = (const float*)d_in[8];
    const float* w_b     = (const float*)d_in[9];
    const float* f1_w    = (const float*)d_in[10];
    const float* f1_b    = (const float*)d_in[11];
    const float* f2_w    = (const float*)d_in[12];
    const float* f2_b    = (const float*)d_in[13];
    const float* dec1_w  = (const float*)d_in[14];
    const float* dec1_b  = (const float*)d_in[15];
    const float* dec3_w  = (const float*)d_in[16];
    const float* dec3_b  = (const float*)d_in[17];

    float* ws = (float*)d_ws;
    const size_t NTW = (size_t)NT * W_;
    float* h     = ws;                 // [NT,128]
    float* g     = ws + 1 * NTW;       // [NT,128]
    float* u     = ws + 2 * NTW;       // [NT,128]
    float* x2    = ws + 3 * NTW;       // [NT,128]
    /* extra  = ws + 4*NTW : second half of hdec256 */
    float* basis = ws + 5 * NTW;       // [8192,64]
    float* stats = basis + (size_t)N_ * K2;   // 1024: mean[512], rstd[512]
    float* sums  = stats + 1024;              // 1024
    float* cc    = sums + 1024;               // 512*64
    float* co    = cc + 512 * K2;             // 512*64
    float* hl256 = g;    // spans g..u   [NT,256]
    float* hdec  = x2;   // spans x2..+  [NT,256]
    float* hl2   = h;    // reuse after decoder gemm1

    // basis tables (recomputed each call: deterministic, no cached state)
    k_basis<<<(N_ * K2 + 255) / 256, 256, 0, stream>>>(basis);

    // ---- GNO encoder ----
    k_enc_edge<<<NT, 128, 0, stream>>>(x, src, enc_e_w, enc_e_b, h);
    k_gemm<<<dim3(NT / 16, 1), 256, 0, stream>>>(h, enc_c_w, enc_c_b,
                                                 nullptr, nullptr, g, 128, 0);
    k_gather<<<NT, 128, 0, stream>>>(g, src, h, 128, 0);

    // ---- 4 FNO blocks ----
    const size_t WWM = (size_t)W_ * W_ * MODES;   // spectral weight stride
    const size_t WW  = (size_t)W_ * W_;           // conv weight stride
    for (int l = 0; l < 4; l++) {
        // instance-norm stats of h
        k_zero<<<4, 256, 0, stream>>>(sums, 1024);
        k_stats_acc<<<dim3(4, 64), 128, 0, stream>>>(h, sums);
        k_stats_fin<<<4, 128, 0, stream>>>(sums, stats);
        // truncated DFT of norm(h): 32 modes
        k_zero<<<128, 256, 0, stream>>>(cc, 512 * K2);
        k_dft<<<dim3(32, 32), 128, 0, stream>>>(h, stats, basis, cc);
        // spectral channel mix (+ fold irfft 1/N and 2x factors)
        k_spectral<<<128, 128, 0, stream>>>(cc, wr + l * WWM, wi + l * WWM, co);
        // truncated iDFT -> g
        k_idft<<<dim3(512, 4), 256, 0, stream>>>(co, basis, g);
        // second instance-norm stats (on g), fused into next GEMM's A-load
        k_zero<<<4, 256, 0, stream>>>(sums, 1024);
        k_stats_acc<<<dim3(4, 64), 128, 0, stream>>>(g, sums);
        k_stats_fin<<<4, 128, 0, stream>>>(sums, stats);
        // u = gelu(norm(g) @ f1^T + b1)
        k_gemm<<<dim3(NT / 16, 1), 256, 0, stream>>>(g, f1_w + l * WW, f1_b + l * W_,
                                                     nullptr, stats, u, 128, 1);
        // x2 = h @ w^T + wb
        k_gemm<<<dim3(NT / 16, 1), 256, 0, stream>>>(h, w_w + l * WW, w_b + l * W_,
                                                     nullptr, nullptr, x2, 128, 0);
        // h = gelu(u @ f2^T + b2 + x2)
        k_gemm<<<dim3(NT / 16, 1), 256, 0, stream>>>(u, f2_w + l * WW, f2_b + l * W_,
                                                     x2, nullptr, h, 128, 1);
    }

    // ---- GNO decoder ----
    k_gemm<<<dim3(NT / 16, 2), 256, 0, stream>>>(h, dec1_w, dec1_b,
                                                 nullptr, nullptr, hl256, 256, 0);
    k_gather<<<NT, 256, 0, stream>>>(hl256, src, hdec, 256, 1);
    k_matvec<<<NT / 8, 256, 0, stream>>>(hdec, dec3_w, dec3_b, hl2);
    k_gather1<<<NT / 256, 256, 0, stream>>>(hl2, src, (float*)d_out);
}